// CRF_1846835937675
// MI455X (gfx1250) — compile-verified
//
#include <hip/hip_runtime.h>
#include <hip/hip_bf16.h>

typedef __attribute__((ext_vector_type(2))) float v2f;
typedef __attribute__((ext_vector_type(8))) float v8f;

#define T2C    128
#define SLENC  1024
#define BT     16     // batch rows per workgroup (WMMA M)
#define ROWPAD 132    // padded row stride for alpha staging (bank-conflict avoidance)

// CRF forward scan. Per workgroup: 16 batch rows, S=1024 sequential steps.
// Each step: new_alpha = log( exp(alpha - m) @ exp(trans)^T ) + m + feat
// GEMM is 16x128x128 in f32 via v_wmma_f32_16x16x4_f32 (8 waves x 32 WMMAs,
// two independent accumulator chains for ILP).
__global__ __launch_bounds__(256) void crf_fwd_kernel(
    const float* __restrict__ feats,
    const float* __restrict__ trans,
    float* __restrict__ out)
{
  __shared__ __align__(16) float Et[32 * 512];      // 64 KB: exp(trans) in B-operand lane order
  __shared__ __align__(16) float Alds[32 * 64];     //  8 KB: exp(alpha-m) in A-operand lane order
  __shared__ __align__(16) float Anew[BT * ROWPAD]; // alpha (log domain) staging
  __shared__ float Mrow[BT];

  const int tid  = threadIdx.x;
  const int wave = tid >> 5;      // wave w owns output tile to = 16w..16w+15
  const int lane = tid & 31;
  const int h    = lane >> 4;
  const int n    = lane & 15;
  const int b0   = blockIdx.x * BT;

  // one-time: Et[kk][w][lane][j] = exp(trans[to=16w+(lane&15)][from=4kk+2*(lane>>4)+j])
  for (int q = 0; q < 64; ++q) {
    int idx = tid + q * 256;
    int kk = idx >> 9;
    int w  = (idx >> 6) & 7;
    int l  = (idx >> 1) & 31;
    int j  = idx & 1;
    int to   = w * 16 + (l & 15);
    int from = kk * 4 + (l >> 4) * 2 + j;
    Et[idx] = __expf(trans[to * T2C + from]);  // exp(MIN_VAL) -> 0 encodes CRF constraints
  }
  // alpha0 = one-hot at from = START_TAG (=127) after exp-shift (m0 = 0)
  for (int q = 0; q < 8; ++q) {
    int f  = tid + q * 256;
    int kk = f >> 6;
    int l  = (f >> 1) & 31;
    int j  = f & 1;
    int K  = kk * 4 + (l >> 4) * 2 + j;
    Alds[f] = (K == (T2C - 1)) ? 1.0f : 0.0f;
  }
  if (tid < BT) Mrow[tid] = 0.0f;
  __syncthreads();

  const size_t rowStride = (size_t)SLENC * T2C;
  const float* fbase = feats + (size_t)(b0 + 8 * h) * rowStride + wave * 16 + n;

  const v2f* Ap = (const v2f*)Alds;              // [kk*32 + lane]
  const v2f* Bp = ((const v2f*)Et) + wave * 32;  // [kk*256 + lane]

  for (int t = 0; t < SLENC; ++t) {
    // issue feat loads early; C/D layout: VGPR i -> row M = i + 8h, col N = n
    float fr[8], mo[8];
    #pragma unroll
    for (int i = 0; i < 8; ++i)
      fr[i] = fbase[(size_t)i * rowStride + (size_t)t * T2C];
    // prefetch next step's feat lines into the WGP cache (speculative; dropped past end)
    #pragma unroll
    for (int i = 0; i < 8; ++i)
      __builtin_prefetch(fbase + (size_t)i * rowStride + (size_t)(t + 1) * T2C, 0, 3);
    #pragma unroll
    for (int i = 0; i < 8; ++i)
      mo[i] = Mrow[8 * h + i];

    // GEMM: acc[b,to] = sum_from exp(alpha-m)[b,from] * exp(trans)[to,from]
    // Two independent accumulator chains (even/odd K chunks) for WMMA ILP.
    v8f acc0 = {0.f, 0.f, 0.f, 0.f, 0.f, 0.f, 0.f, 0.f};
    v8f acc1 = {0.f, 0.f, 0.f, 0.f, 0.f, 0.f, 0.f, 0.f};
    #pragma unroll
    for (int kk = 0; kk < 32; kk += 2) {
      v2f a0 = Ap[(kk + 0) * 32 + lane];
      v2f b0 = Bp[(kk + 0) * 256 + lane];
      v2f a1 = Ap[(kk + 1) * 32 + lane];
      v2f b1 = Bp[(kk + 1) * 256 + lane];
      acc0 = __builtin_amdgcn_wmma_f32_16x16x4_f32(false, a0, false, b0,
                                                   (short)0, acc0, false, false);
      acc1 = __builtin_amdgcn_wmma_f32_16x16x4_f32(false, a1, false, b1,
                                                   (short)0, acc1, false, false);
    }
    v8f acc = acc0 + acc1;

    // back to log domain; stage alpha_{t+1}
    #pragma unroll
    for (int i = 0; i < 8; ++i) {
      float v = __logf(acc[i]) + mo[i] + fr[i];  // log(0) = -inf ok: exp -> 0 next step
      Anew[(i + 8 * h) * ROWPAD + wave * 16 + n] = v;
    }
    __syncthreads();

    // per-row max, exp, repack into WMMA A layout (16 threads per row)
    {
      int r = tid >> 4, g = tid & 15;
      const float4* rowp = (const float4*)(Anew + r * ROWPAD);
      float4 v0 = rowp[g * 2 + 0];
      float4 v1 = rowp[g * 2 + 1];
      float vv[8] = {v0.x, v0.y, v0.z, v0.w, v1.x, v1.y, v1.z, v1.w};
      float mx = vv[0];
      #pragma unroll
      for (int i2 = 1; i2 < 8; ++i2) mx = fmaxf(mx, vv[i2]);
      #pragma unroll
      for (int off = 1; off < 16; off <<= 1)      // 16-lane butterfly (wave32-safe)
        mx = fmaxf(mx, __shfl_xor(mx, off, 32));
      #pragma unroll
      for (int p = 0; p < 4; ++p) {
        int c  = g * 8 + p * 2;                   // pair (c, c+1) shares (kk, half)
        int kk = c >> 2;
        int hh = (c >> 1) & 1;
        float2 e;
        e.x = __expf(vv[p * 2 + 0] - mx);
        e.y = __expf(vv[p * 2 + 1] - mx);
        *(float2*)(Alds + kk * 64 + (16 * hh + r) * 2) = e;
      }
      if (g == 0) Mrow[r] = mx;
    }
    __syncthreads();
  }

  // final: out[b] = logsumexp_from( alpha_S[b,from] + trans[END_TAG=126, from] )
  {
    int r = tid >> 4, g = tid & 15;
    float vv[8];
    #pragma unroll
    for (int j2 = 0; j2 < 8; ++j2)
      vv[j2] = Anew[r * ROWPAD + g * 8 + j2] + trans[(T2C - 2) * T2C + g * 8 + j2];
    float mx = vv[0];
    #pragma unroll
    for (int i2 = 1; i2 < 8; ++i2) mx = fmaxf(mx, vv[i2]);
    #pragma unroll
    for (int off = 1; off < 16; off <<= 1)
      mx = fmaxf(mx, __shfl_xor(mx, off, 32));
    float s = 0.f;
    #pragma unroll
    for (int j2 = 0; j2 < 8; ++j2) s += __expf(vv[j2] - mx);
    #pragma unroll
    for (int off = 1; off < 16; off <<= 1)
      s += __shfl_xor(s, off, 32);
    if (g == 0) out[b0 + r] = __logf(s) + mx;
  }
}

extern "C" void kernel_launch(void* const* d_in, const int* in_sizes, int n_in,
                              void* d_out, int out_size, void* d_ws, size_t ws_size,
                              hipStream_t stream) {
  const float* feats = (const float*)d_in[0];
  const float* trans = (const float*)d_in[1];
  float* out = (float*)d_out;
  int B = in_sizes[0] / (SLENC * T2C);   // 256
  crf_fwd_kernel<<<dim3(B / BT), dim3(256), 0, stream>>>(feats, trans, out);
}